// SelfAttention_33251636805769
// MI455X (gfx1250) — compile-verified
//
#include <hip/hip_runtime.h>
#include <math.h>

// Problem constants (fixed by the reference)
#define SB   2048          // sequence length S
#define DM   1024          // model dim D
#define NH   16            // heads
#define HD   64            // head dim
#define HD2  32            // HD/2
#define SCALE 0.125f       // 1/sqrt(64)
#define EPS  1e-6f

typedef __attribute__((ext_vector_type(2))) float v2f;
typedef __attribute__((ext_vector_type(8))) float v8f;

static __device__ __forceinline__ v8f wmma_f32(v2f a, v2f b, v8f c) {
    // V_WMMA_F32_16X16X4_F32 : D = A(16x4,f32) * B(4x16,f32) + C(16x16,f32)
    return __builtin_amdgcn_wmma_f32_16x16x4_f32(
        /*neg_a=*/false, a, /*neg_b=*/false, b,
        /*c_mod=*/(short)0, c, /*reuse_a=*/false, /*reuse_b=*/false);
}

// ---------------------------------------------------------------------------
// Kernel 1/3: out[s, :] = rope(rmsnorm(x @ W^T + bias)) (mode 1)
//             out[s, :] = x @ W^T + bias                (mode 0)
// One workgroup = 16 rows x 1024 cols. 8 waves; wave w owns cols [w*128, w*128+128).
// A-layout (16x4 f32): lane l -> row M = l&15, K pair offset 2*(l>>4).
// B-layout (4x16 f32): lane l -> col N = l&15, K pair offset 2*(l>>4).
// C-layout: vgpr j, lane l -> row j + 8*(l>>4), col l&15.
// ---------------------------------------------------------------------------
__global__ __launch_bounds__(256) void gemm16_kernel(
    const float* __restrict__ X,     // [SB, DM]
    const float* __restrict__ W,     // [DM, DM] row-major, out = X @ W^T
    const float* __restrict__ bias,  // [DM]
    const float* __restrict__ g,     // [DM] rms gain (mode 1)
    const float* __restrict__ fc,    // [SB, HD2] cos
    const float* __restrict__ fs,    // [SB, HD2] sin
    float* __restrict__ out,         // [SB, DM]
    int doNormRope)
{
    __shared__ float tile[16][DM + 4];   // 16 x 1028 f32 (~66 KB of the 320 KB WGP LDS)
    __shared__ float redbuf[16][17];
    __shared__ float rrms[16];

    const int tid  = threadIdx.x;
    const int wave = tid >> 5;
    const int lane = tid & 31;
    const int half = lane >> 4;
    const int l16  = lane & 15;
    const int rowBase  = blockIdx.x * 16;
    const int nColBase = wave * 128;

    v8f acc[8];
#pragma unroll
    for (int t = 0; t < 8; ++t) acc[t] = (v8f){0,0,0,0,0,0,0,0};

    const float* xrow = X + (size_t)(rowBase + l16) * DM + 2 * half;

    for (int k = 0; k < DM; k += 4) {
        v2f a = *(const v2f*)(xrow + k);
#pragma unroll
        for (int t = 0; t < 8; ++t) {
            const float* wp = W + (size_t)(nColBase + t * 16 + l16) * DM + k + 2 * half;
            v2f b = *(const v2f*)wp;
            acc[t] = wmma_f32(a, b, acc[t]);
        }
    }

    // spill C-layout accumulators (+bias) into the LDS row tile
#pragma unroll
    for (int t = 0; t < 8; ++t) {
        int col = nColBase + t * 16 + l16;
        float bv = bias[col];
#pragma unroll
        for (int j = 0; j < 8; ++j) {
            int row = j + 8 * half;
            tile[row][col] = acc[t][j] + bv;
        }
    }
    __syncthreads();

    if (doNormRope) {
        // sum of squares per row: 16 threads per row, strided
        {
            int r  = tid & 15;
            int c0 = tid >> 4;
            float s = 0.f;
            for (int c = c0; c < DM; c += 16) { float v = tile[r][c]; s += v * v; }
            redbuf[r][c0] = s;
        }
        __syncthreads();
        if (tid < 16) {
            float s = 0.f;
#pragma unroll
            for (int c = 0; c < 16; ++c) s += redbuf[tid][c];
            rrms[tid] = rsqrtf(s * (1.0f / (float)DM) + EPS);
        }
        __syncthreads();
    }

    // apply norm + rope (or pass-through) and store; 16 rows x 512 pairs
    for (int p = tid; p < 16 * (DM / 2); p += 256) {
        int r    = p >> 9;          // / 512
        int cp   = p & 511;
        int dcol = cp * 2;
        int srow = rowBase + r;
        float re = tile[r][dcol];
        float im = tile[r][dcol + 1];
        if (doNormRope) {
            float rr = rrms[r];
            re *= rr * g[dcol];
            im *= rr * g[dcol + 1];
            int   pp = cp & (HD2 - 1);           // pair index within head
            float c  = fc[srow * HD2 + pp];
            float sn = fs[srow * HD2 + pp];
            float ore = re * c - im * sn;
            float oim = re * sn + im * c;
            re = ore; im = oim;
        }
        out[(size_t)srow * DM + dcol]     = re;
        out[(size_t)srow * DM + dcol + 1] = im;
    }
}

// ---------------------------------------------------------------------------
// Kernel 2: non-causal flash attention. One wave per (head, 16-query tile).
// ---------------------------------------------------------------------------
__global__ __launch_bounds__(32) void attn_kernel(
    const float* __restrict__ Q,   // [SB, DM]
    const float* __restrict__ K,
    const float* __restrict__ V,
    float* __restrict__ O)         // [SB, DM]
{
    __shared__ float P[16][17];    // C-layout -> A-layout bounce buffer

    const int lane = threadIdx.x & 31;
    const int half = lane >> 4;
    const int l16  = lane & 15;
    const int qt    = blockIdx.x;        // query tile
    const int h     = blockIdx.y;        // head
    const int qBase = qt * 16;
    const int dBase = h * HD;

    // preload Q tile in A-operand layout: 16 chunks of K=4
    v2f qa[16];
    {
        const float* qp = Q + (size_t)(qBase + l16) * DM + dBase + 2 * half;
#pragma unroll
        for (int kk = 0; kk < 16; ++kk) qa[kk] = *(const v2f*)(qp + kk * 4);
    }

    v8f o[4];
#pragma unroll
    for (int nt = 0; nt < 4; ++nt) o[nt] = (v8f){0,0,0,0,0,0,0,0};
    float m[8], lsum[8];
#pragma unroll
    for (int j = 0; j < 8; ++j) { m[j] = -INFINITY; lsum[j] = 0.f; }

    for (int t = 0; t < SB / 16; ++t) {
        const int kBase = t * 16;

        // S = Q * K^T  (16x16), K=64 in 16 WMMA steps
        v8f s = (v8f){0,0,0,0,0,0,0,0};
        const float* kp = K + (size_t)(kBase + l16) * DM + dBase + 2 * half;
#pragma unroll
        for (int kk = 0; kk < 16; ++kk) {
            v2f b = *(const v2f*)(kp + kk * 4);
            s = wmma_f32(qa[kk], b, s);
        }

        // online softmax; row j+8*half lives in the 16 lanes of this half
#pragma unroll
        for (int j = 0; j < 8; ++j) {
            float sv = s[j] * SCALE;
            float mx = sv;
            mx = fmaxf(mx, __shfl_xor(mx, 1, 32));
            mx = fmaxf(mx, __shfl_xor(mx, 2, 32));
            mx = fmaxf(mx, __shfl_xor(mx, 4, 32));
            mx = fmaxf(mx, __shfl_xor(mx, 8, 32));
            float nm   = fmaxf(m[j], mx);
            float corr = __expf(m[j] - nm);
            float pv   = __expf(sv - nm);
            float ps = pv;
            ps += __shfl_xor(ps, 1, 32);
            ps += __shfl_xor(ps, 2, 32);
            ps += __shfl_xor(ps, 4, 32);
            ps += __shfl_xor(ps, 8, 32);
            lsum[j] = lsum[j] * corr + ps;
            m[j] = nm;
            o[0][j] *= corr; o[1][j] *= corr; o[2][j] *= corr; o[3][j] *= corr;
            P[j + 8 * half][l16] = pv;       // C-layout position
        }
        __syncthreads();

        // O += P * V  (16x16 @ 16x64) : 4 K-chunks x 4 N-tiles
#pragma unroll
        for (int kc = 0; kc < 4; ++kc) {
            v2f pa;
            pa.x = P[l16][kc * 4 + 2 * half];
            pa.y = P[l16][kc * 4 + 2 * half + 1];
#pragma unroll
            for (int nt = 0; nt < 4; ++nt) {
                const float* vp = V + (size_t)(kBase + kc * 4 + 2 * half) * DM
                                    + dBase + nt * 16 + l16;
                v2f b; b.x = vp[0]; b.y = vp[DM];
                o[nt] = wmma_f32(pa, b, o[nt]);
            }
        }
        __syncthreads();
    }

    // normalize and store
#pragma unroll
    for (int j = 0; j < 8; ++j) {
        float inv = 1.0f / lsum[j];
        int row = qBase + j + 8 * half;
#pragma unroll
        for (int nt = 0; nt < 4; ++nt) {
            O[(size_t)row * DM + dBase + nt * 16 + l16] = o[nt][j] * inv;
        }
    }
}

// ---------------------------------------------------------------------------
extern "C" void kernel_launch(void* const* d_in, const int* in_sizes, int n_in,
                              void* d_out, int out_size, void* d_ws, size_t ws_size,
                              hipStream_t stream) {
    const float* x  = (const float*)d_in[0];
    const float* fc = (const float*)d_in[1];
    const float* fs = (const float*)d_in[2];
    const float* wq = (const float*)d_in[3];
    const float* bq = (const float*)d_in[4];
    const float* wk = (const float*)d_in[5];
    const float* bk = (const float*)d_in[6];
    const float* wv = (const float*)d_in[7];
    const float* bv = (const float*)d_in[8];
    const float* wo = (const float*)d_in[9];
    const float* bo = (const float*)d_in[10];
    const float* gq = (const float*)d_in[11];
    const float* gk = (const float*)d_in[12];
    float* out = (float*)d_out;

    const size_t mat = (size_t)SB * DM;      // 2M floats = 8 MB each
    float* Qw = (float*)d_ws;
    float* Kw = Qw + mat;
    float* Vw = Kw + mat;
    float* Aw = Vw + mat;

    dim3 gblk(SB / 16);   // 128 row-tiles
    gemm16_kernel<<<gblk, 256, 0, stream>>>(x, wq, bq, gq, fc, fs, Qw, 1);
    gemm16_kernel<<<gblk, 256, 0, stream>>>(x, wk, bk, gk, fc, fs, Kw, 1);
    gemm16_kernel<<<gblk, 256, 0, stream>>>(x, wv, bv, gq, fc, fs, Vw, 0);
    attn_kernel<<<dim3(SB / 16, NH), 32, 0, stream>>>(Qw, Kw, Vw, Aw);
    gemm16_kernel<<<gblk, 256, 0, stream>>>(Aw, wo, bo, gq, fc, fs, out, 0);
}